// GAT_41601053229135
// MI455X (gfx1250) — compile-verified
//
#include <hip/hip_runtime.h>
#include <hip/hip_bf16.h>

#define HEADS 4
#define OUTW 64
#define FEAT 256           // HEADS*OUT
#define NEG_SLOPE 0.2f
#define BN_EPS 1e-5f

typedef __attribute__((ext_vector_type(2))) float v2f;
typedef __attribute__((ext_vector_type(8))) float v8f;

// ---------------------------------------------------------------------------
// fp32 WMMA GEMM: C[M,N] = A[M,K] @ B[K,N], row-major. N % 64 == 0,
// K % 32 == 0, M % 16 == 0.
// Block = 8 waves -> 128x64 C tile. Per 32-deep K chunk the block stages the
// 32x64 B panel into LDS *pair-swizzled*: Bs[(k>>1)*128 + c*2 + (k&1)], so
// the two K-values a lane needs for a fragment are one aligned b64 read.
// Each wave drives 4 accumulators (16x64) with v_wmma_f32_16x16x4_f32,
// reusing its A fragment across the 4 N-subtiles.
// Frag layouts (ISA 7.12.2, 32-bit): lane l<16 holds K=k..k+1 of row l,
// lane l>=16 holds K=k+2..k+3.  C/D: VGPR v -> row v (lanes 0-15) / v+8.
// ---------------------------------------------------------------------------
__global__ __launch_bounds__(256) void gemm16_wmma(
    const float* __restrict__ A, const float* __restrict__ B,
    float* __restrict__ C, int M, int N, int K) {
  __shared__ __align__(16) float Bs[32 * 64];
  const int wave = threadIdx.x >> 5;
  const int lane = threadIdx.x & 31;
  const int mt = blockIdx.x * 8 + wave;          // 16-row tile index
  const bool active = (mt << 4) < M;             // wave-uniform guard
  const int nBase = blockIdx.y * 64;
  const int half = lane >> 4;                    // 0 or 1
  const int l15 = lane & 15;
  const int row = (mt << 4) + l15;               // A row this lane feeds
  const int koff = half << 1;                    // 0 or 2

  v8f acc0 = {}, acc1 = {}, acc2 = {}, acc3 = {};
  const float* ap = active ? (A + (size_t)row * K + koff) : A;

  for (int k0 = 0; k0 < K; k0 += 32) {
    __syncthreads();   // protect Bs against readers of previous chunk
    // ---- cooperative stage of B[k0..k0+31][nBase..nBase+63], pair-swizzled:
    //      Bs[p*128 + c*2 + 0/1] = B[k0+2p (+1)][nBase+c]
#pragma unroll
    for (int i = 0; i < 4; ++i) {
      int flat = i * 256 + threadIdx.x;          // 0..1023 (16 pairs x 64 cols)
      int p = flat >> 6;
      int c = flat & 63;
      v2f t;
      t.x = B[(size_t)(k0 + 2 * p) * N + nBase + c];
      t.y = B[(size_t)(k0 + 2 * p + 1) * N + nBase + c];
      *(v2f*)&Bs[p * 128 + c * 2] = t;           // ds_store_b64, conflict-free
    }
    __syncthreads();
    // speculative prefetch of next A chunk (global_prefetch_b8)
    __builtin_prefetch(ap + k0 + 32, 0, 1);
    if (active) {
#pragma unroll
      for (int kk = 0; kk < 32; kk += 4) {
        v2f a;
        a.x = ap[k0 + kk];                       // one global_load_b64 per lane
        a.y = ap[k0 + kk + 1];
        // pair index for rows (kk+koff, kk+koff+1); koff even so exact pair
        const float* bs = &Bs[((kk + koff) >> 1) * 128 + l15 * 2];
        v2f b0 = *(const v2f*)(bs + 0);          // ds_load_b64 x4 (2addr-merged)
        v2f b1 = *(const v2f*)(bs + 32);
        v2f b2 = *(const v2f*)(bs + 64);
        v2f b3 = *(const v2f*)(bs + 96);
        acc0 = __builtin_amdgcn_wmma_f32_16x16x4_f32(false, a, false, b0,
                                                     (short)0, acc0, false, false);
        acc1 = __builtin_amdgcn_wmma_f32_16x16x4_f32(false, a, false, b1,
                                                     (short)0, acc1, false, false);
        acc2 = __builtin_amdgcn_wmma_f32_16x16x4_f32(false, a, false, b2,
                                                     (short)0, acc2, false, false);
        acc3 = __builtin_amdgcn_wmma_f32_16x16x4_f32(false, a, false, b3,
                                                     (short)0, acc3, false, false);
      }
    }
  }
  if (active) {
    const int crow = (mt << 4) + (half << 3);
    float* cp = C + (size_t)crow * N + nBase + l15;
#pragma unroll
    for (int v = 0; v < 8; ++v) {
      cp[(size_t)v * N + 0]  = acc0[v];
      cp[(size_t)v * N + 16] = acc1[v];
      cp[(size_t)v * N + 32] = acc2[v];
      cp[(size_t)v * N + 48] = acc3[v];
    }
  }
}

// ---------------------------------------------------------------------------
// attention projections: one wave per (node, head); coalesced 64-float row
// read + shuffle reduction.  asrc[n,h] = h[n,h,:] . a_src[h,:]  (and adst)
// ---------------------------------------------------------------------------
__global__ void attn_scores(const float* __restrict__ h,
                            const float* __restrict__ a_src,
                            const float* __restrict__ a_dst,
                            float* __restrict__ asrc, float* __restrict__ adst,
                            int N) {
  int w = (blockIdx.x * blockDim.x + threadIdx.x) >> 5;  // (n*HEADS+hd)
  int lane = threadIdx.x & 31;
  if (w >= N * HEADS) return;                            // wave-uniform
  int n = w >> 2;
  int hd = w & 3;
  const float* hp = h + (size_t)n * FEAT + hd * OUTW;
  const float* as = a_src + hd * OUTW;
  const float* ad = a_dst + hd * OUTW;
  float v0 = hp[lane];
  float v1 = hp[lane + 32];
  float ss = v0 * as[lane] + v1 * as[lane + 32];
  float sd = v0 * ad[lane] + v1 * ad[lane + 32];
#pragma unroll
  for (int off = 16; off; off >>= 1) {
    ss += __shfl_down(ss, off, 32);
    sd += __shfl_down(sd, off, 32);
  }
  if (lane == 0) {
    asrc[w] = ss;
    adst[w] = sd;
  }
}

__global__ void fill_f32(float* __restrict__ p, float v, long long n) {
  long long i = (long long)blockIdx.x * blockDim.x + threadIdx.x;
  long long st = (long long)gridDim.x * blockDim.x;
  for (; i < n; i += st) p[i] = v;
}

__device__ __forceinline__ void atomicMaxF(float* addr, float val) {
  int* ia = (int*)addr;
  int cur = __float_as_int(*addr);
  while (__int_as_float(cur) < val) {
    int old = atomicCAS(ia, cur, __float_as_int(val));
    if (old == cur) break;
    cur = old;
  }
}

// logits = leaky_relu(asrc[src] + adst[dst]); running segment max over dst
__global__ void edge_logits_max(const int* __restrict__ src,
                                const int* __restrict__ dst,
                                const float* __restrict__ asrc,
                                const float* __restrict__ adst,
                                float* __restrict__ elog,
                                float* __restrict__ smax, int M) {
  int idx = blockIdx.x * blockDim.x + threadIdx.x;
  if (idx >= M * HEADS) return;
  int e = idx >> 2;
  int hd = idx & 3;
  int s = src[e];
  int d = dst[e];
  float v = asrc[s * HEADS + hd] + adst[d * HEADS + hd];
  v = (v > 0.f) ? v : NEG_SLOPE * v;
  elog[idx] = v;
  atomicMaxF(&smax[d * HEADS + hd], v);
}

// e = exp(logit - max[dst]); running segment sum over dst
__global__ void edge_exp_sum(const int* __restrict__ dst,
                             float* __restrict__ elog,
                             const float* __restrict__ smax,
                             float* __restrict__ sden, int M) {
  int idx = blockIdx.x * blockDim.x + threadIdx.x;
  if (idx >= M * HEADS) return;
  int e = idx >> 2;
  int hd = idx & 3;
  int d = dst[e];
  float ex = __expf(elog[idx] - smax[d * HEADS + hd]);
  elog[idx] = ex;
  atomicAdd(&sden[d * HEADS + hd], ex);
}

// out[dst] += h[src] * (e/denom[dst]) : one block of 256 threads per edge
__global__ __launch_bounds__(256) void scatter_msg(
    const int* __restrict__ src, const int* __restrict__ dst,
    const float* __restrict__ h, const float* __restrict__ elog,
    const float* __restrict__ sden, float* __restrict__ out) {
  int e = blockIdx.x;
  int f = threadIdx.x;       // 0..255
  int hd = f >> 6;
  int s = src[e];
  int d = dst[e];
  float alpha = elog[e * HEADS + hd] / sden[d * HEADS + hd];
  atomicAdd(&out[(size_t)d * FEAT + f], h[(size_t)s * FEAT + f] * alpha);
}

// h = relu(bn(h + bias))
__global__ void bias_bn_relu(float* __restrict__ h, const float* __restrict__ b,
                             const float* __restrict__ g,
                             const float* __restrict__ be,
                             const float* __restrict__ mean,
                             const float* __restrict__ var, long long n) {
  long long idx = (long long)blockIdx.x * blockDim.x + threadIdx.x;
  if (idx >= n) return;
  int c = (int)(idx & (FEAT - 1));
  float v = h[idx] + b[c];
  v = (v - mean[c]) * rsqrtf(var[c] + BN_EPS) * g[c] + be[c];
  h[idx] = v > 0.f ? v : 0.f;
}

// concat=False: mean over heads + bias
__global__ void head_mean_bias(const float* __restrict__ hin,
                               const float* __restrict__ b2,
                               float* __restrict__ h2, int N) {
  int idx = blockIdx.x * blockDim.x + threadIdx.x;
  if (idx >= N * OUTW) return;
  int n = idx >> 6;
  int o = idx & 63;
  const float* p = hin + (size_t)n * FEAT + o;
  float s = (p[0] + p[OUTW] + p[2 * OUTW] + p[3 * OUTW]) * 0.25f + b2[o];
  h2[idx] = s;
}

// incidence aggregation: agg[rows[e]] += h2[e]; agg[cols[e]] += h2[e] if r!=c
__global__ void final_scatter(const int* __restrict__ rows,
                              const int* __restrict__ cols,
                              const float* __restrict__ h2,
                              float* __restrict__ agg, float* __restrict__ deg,
                              int E) {
  int idx = blockIdx.x * blockDim.x + threadIdx.x;
  if (idx >= E * OUTW) return;
  int e = idx >> 6;
  int f = idx & 63;
  int r = rows[e];
  int c = cols[e];
  float v = h2[idx];
  atomicAdd(&agg[(size_t)r * OUTW + f], v);
  if (r != c) atomicAdd(&agg[(size_t)c * OUTW + f], v);
  if (f == 0) {
    atomicAdd(&deg[r], 1.0f);
    if (r != c) atomicAdd(&deg[c], 1.0f);
  }
}

__global__ void normalize_deg(float* __restrict__ agg,
                              const float* __restrict__ deg, int Nn) {
  int idx = blockIdx.x * blockDim.x + threadIdx.x;
  if (idx >= Nn * OUTW) return;
  agg[idx] = agg[idx] / deg[idx >> 6];
}

// ---------------------------------------------------------------------------
extern "C" void kernel_launch(void* const* d_in, const int* in_sizes, int n_in,
                              void* d_out, int out_size, void* d_ws,
                              size_t ws_size, hipStream_t stream) {
  const float* x = (const float*)d_in[0];
  const int* ei = (const int*)d_in[1];
  const int* rows = (const int*)d_in[2];
  const int* cols = (const int*)d_in[3];
  const float* W0 = (const float*)d_in[5];
  const float* as0 = (const float*)d_in[6];
  const float* ad0 = (const float*)d_in[7];
  const float* b0 = (const float*)d_in[8];
  const float* g0 = (const float*)d_in[9];
  const float* be0 = (const float*)d_in[10];
  const float* m0 = (const float*)d_in[11];
  const float* v0 = (const float*)d_in[12];
  const float* W1 = (const float*)d_in[13];
  const float* as1 = (const float*)d_in[14];
  const float* ad1 = (const float*)d_in[15];
  const float* b1 = (const float*)d_in[16];
  const float* g1 = (const float*)d_in[17];
  const float* be1 = (const float*)d_in[18];
  const float* m1 = (const float*)d_in[19];
  const float* v1 = (const float*)d_in[20];
  const float* W2 = (const float*)d_in[21];
  const float* as2 = (const float*)d_in[22];
  const float* ad2 = (const float*)d_in[23];
  const float* b2 = (const float*)d_in[24];
  const float* Wout = (const float*)d_in[25];

  const int E = in_sizes[2];                // 100000 feature nodes
  const int Fin = in_sizes[0] / E;          // 128
  const int M = in_sizes[1] / 2;            // 800000 edges
  const int Nn = out_size / OUTW;           // 50000 output nodes
  const int* src = ei;
  const int* dst = ei + M;

  // workspace layout (floats)
  float* ws = (float*)d_ws;
  size_t off = 0;
  float* hA = ws + off;   off += (size_t)E * FEAT;
  float* hB = ws + off;   off += (size_t)E * FEAT;
  float* asrc = ws + off; off += (size_t)E * HEADS;
  float* adst = ws + off; off += (size_t)E * HEADS;
  float* smax = ws + off; off += (size_t)E * HEADS;
  float* sden = ws + off; off += (size_t)E * HEADS;
  float* elog = ws + off; off += (size_t)M * HEADS;
  float* h2 = ws + off;   off += (size_t)E * OUTW;
  float* agg = ws + off;  off += (size_t)Nn * OUTW;
  float* deg = ws + off;  off += (size_t)Nn;
  (void)ws_size; (void)n_in;

  auto cdiv = [](long long a, long long b) { return (int)((a + b - 1) / b); };

  auto gemm = [&](const float* A, const float* B, float* C, int Mm, int Nc,
                  int K) {
    dim3 grid(cdiv(Mm >> 4, 8), Nc >> 6);
    gemm16_wmma<<<grid, 256, 0, stream>>>(A, B, C, Mm, Nc, K);
  };

  // one GAT layer edge phase: h (features) in hA, scatter result into hB
  auto edge_phase = [&](const float* a_s, const float* a_d) {
    attn_scores<<<cdiv((long long)E * HEADS * 32, 256), 256, 0, stream>>>(
        hA, a_s, a_d, asrc, adst, E);
    fill_f32<<<2048, 256, 0, stream>>>(smax, -INFINITY, (long long)E * HEADS);
    fill_f32<<<2048, 256, 0, stream>>>(sden, 0.f, (long long)E * HEADS);
    edge_logits_max<<<cdiv((long long)M * HEADS, 256), 256, 0, stream>>>(
        src, dst, asrc, adst, elog, smax, M);
    edge_exp_sum<<<cdiv((long long)M * HEADS, 256), 256, 0, stream>>>(
        dst, elog, smax, sden, M);
    fill_f32<<<4096, 256, 0, stream>>>(hB, 0.f, (long long)E * FEAT);
    scatter_msg<<<M, 256, 0, stream>>>(src, dst, hA, elog, sden, hB);
  };

  // ---- layer 0: x[ E,128 ] -> hB[ E,256 ], then BN+ReLU in place
  gemm(x, W0, hA, E, FEAT, Fin);
  edge_phase(as0, ad0);
  bias_bn_relu<<<cdiv((long long)E * FEAT, 256), 256, 0, stream>>>(
      hB, b0, g0, be0, m0, v0, (long long)E * FEAT);

  // ---- layer 1: hB -> hB
  gemm(hB, W1, hA, E, FEAT, FEAT);
  edge_phase(as1, ad1);
  bias_bn_relu<<<cdiv((long long)E * FEAT, 256), 256, 0, stream>>>(
      hB, b1, g1, be1, m1, v1, (long long)E * FEAT);

  // ---- layer 2: hB -> head-mean -> h2[ E,64 ]
  gemm(hB, W2, hA, E, FEAT, FEAT);
  edge_phase(as2, ad2);
  head_mean_bias<<<cdiv((long long)E * OUTW, 256), 256, 0, stream>>>(
      hB, b2, h2, E);

  // ---- incidence aggregation + degree normalize + final projection
  fill_f32<<<2048, 256, 0, stream>>>(agg, 0.f, (long long)Nn * OUTW);
  fill_f32<<<256, 256, 0, stream>>>(deg, 0.f, (long long)Nn);
  final_scatter<<<cdiv((long long)E * OUTW, 256), 256, 0, stream>>>(
      rows, cols, h2, agg, deg, E);
  normalize_deg<<<cdiv((long long)Nn * OUTW, 256), 256, 0, stream>>>(
      agg, deg, Nn);
  gemm(agg, Wout, (float*)d_out, Nn, OUTW, OUTW);
}